// NexToU_Encoder_16183436772199
// MI455X (gfx1250) — compile-verified
//
#include <hip/hip_runtime.h>

#define B_   2
#define C_   128
#define C2_  256
#define S_   16
#define H_   48
#define W_   48
#define N_   (S_*H_*W_)       /* 36864 */
#define SP_  8
#define HP_  24
#define WP_  24
#define NP_  (SP_*HP_*WP_)    /* 4608 */
#define KNN_ 9
#define EPS_ 1e-5f
#define SLOPE_ 0.2f

typedef __attribute__((ext_vector_type(2))) float v2f;
typedef __attribute__((ext_vector_type(8))) float v8f;

// ---------------------------------------------------------------------------
// CDNA5 async global->LDS copy (ASYNCcnt-tracked) + wait.
// LDS destination address = low 32 bits of the flat shared pointer
// (ISA 10.2: LDS aperture keeps the byte offset in addr[31:0]).
// ---------------------------------------------------------------------------
__device__ __forceinline__ void g2l_async_b128(float* lds_dst, const float* gsrc)
{
  unsigned lo = (unsigned)(size_t)lds_dst;
  asm volatile("global_load_async_to_lds_b128 %0, %1, off"
               :: "v"(lo), "v"(gsrc) : "memory");
}
__device__ __forceinline__ void wait_async0()
{
  asm volatile("s_wait_asynccnt 0x0" ::: "memory");
}

// ---------------------------------------------------------------------------
// Dense WMMA GEMM with LDS-staged X panel:
//   Y[b] = W (O x Kd) * X[b] (Kd x Ncols) + bias
// One 256-thread block (8 waves) per 16-column strip. The Kd x 16 X panel is
// async-copied into LDS once (exactly 1x X traffic), B-fragments are read from
// LDS, A-fragments stream from the (L2-resident) weight matrix.
// ---------------------------------------------------------------------------
template<int O, int Kd>
__global__ __launch_bounds__(256)
void k_gemm_panel(const float* __restrict__ Wm, const float* __restrict__ bias,
                  const float* __restrict__ X, float* __restrict__ Y, int Ncols)
{
  __shared__ __align__(16) float panel[Kd * 16];
  int nt_total = Ncols >> 4;
  int b  = blockIdx.x / nt_total;
  int nt = blockIdx.x % nt_total;
  const float* Xb = X + (size_t)b * Kd * Ncols + nt * 16;

  // stage panel: Kd rows of 16 floats (64B) = Kd*4 b128 transfers
  for (int i = threadIdx.x; i < Kd * 4; i += 256) {
    int row = i >> 2, cg = i & 3;
    g2l_async_b128(&panel[row * 16 + cg * 4], Xb + (size_t)row * Ncols + cg * 4);
  }
  wait_async0();
  __syncthreads();

  int wave = threadIdx.x >> 5;
  int lane = threadIdx.x & 31;
  int half = lane >> 4, l15 = lane & 15;
  float* Yb = Y + (size_t)b * O * Ncols + nt * 16;

  for (int mt = wave; mt < O / 16; mt += 8) {
    int mrow = mt * 16 + l15;
    v8f acc = {};
    for (int k = 0; k < Kd; k += 4) {
      int kk = k + half * 2;                    // lanes16-31 hold upper K half
      v2f a, bf;
      a.x  = Wm[(size_t)mrow * Kd + kk];
      a.y  = Wm[(size_t)mrow * Kd + kk + 1];
      bf.x = panel[kk * 16 + l15];
      bf.y = panel[(kk + 1) * 16 + l15];
      acc = __builtin_amdgcn_wmma_f32_16x16x4_f32(false, a, false, bf,
                                                  (short)0, acc, false, false);
    }
#pragma unroll
    for (int rr = 0; rr < 8; rr++) {
      int m = mt * 16 + rr + half * 8;          // C/D layout
      Yb[(size_t)m * Ncols + l15] = acc[rr] + bias[m];
    }
  }
}

// ---------------------------------------------------------------------------
// fc2 GEMM with fused MaxUnpool3d, panel style.
// One 256-thread block per 16-column strip (2 pooled nodes x 8 offsets).
// Staging: read y2[ch,p] / idx[ch mod C, p] once and expand the masked
// 256x16 B-panel into LDS with branch-free selects.  Compute: 8 waves, one
// 16x16 WMMA tile each; output column (p,q) remapped to voxel (s,h,w).
// ---------------------------------------------------------------------------
__global__ __launch_bounds__(256)
void k_gemm_fc2(const float* __restrict__ Wm, const float* __restrict__ bias,
                const float* __restrict__ y2, const int* __restrict__ idxp,
                float* __restrict__ Yout)
{
  __shared__ __align__(16) float panel[C2_ * 16];
  const int nt_total = N_ / 16;
  int b  = blockIdx.x / nt_total;
  int nt = blockIdx.x % nt_total;
  const float* Y2b = y2 + (size_t)b * C2_ * NP_;
  const int* Ib = idxp + (size_t)b * C_ * NP_;
  int p0 = nt * 2;                              // strip covers nodes p0, p0+1

  for (int i = threadIdx.x; i < C2_ * 2; i += 256) {
    int ch = i >> 1, pi = i & 1;
    int p = p0 + pi;
    float v = Y2b[(size_t)ch * NP_ + p];
    int iv = Ib[(size_t)(ch < C_ ? ch : ch - C_) * NP_ + p];
    float* dst = &panel[ch * 16 + pi * 8];
#pragma unroll
    for (int q = 0; q < 8; q++) dst[q] = (q == iv) ? v : 0.0f;
  }
  __syncthreads();

  int wave = threadIdx.x >> 5;                  // 8 waves == 8 m-tiles (O=128)
  int lane = threadIdx.x & 31;
  int half = lane >> 4, l15 = lane & 15;
  int mt = wave;
  int mrow = mt * 16 + l15;
  v8f acc = {};
  for (int k = 0; k < C2_; k += 4) {
    int kk = k + half * 2;
    v2f a, bf;
    a.x  = Wm[(size_t)mrow * C2_ + kk];
    a.y  = Wm[(size_t)mrow * C2_ + kk + 1];
    bf.x = panel[kk * 16 + l15];
    bf.y = panel[(kk + 1) * 16 + l15];
    acc = __builtin_amdgcn_wmma_f32_16x16x4_f32(false, a, false, bf,
                                                (short)0, acc, false, false);
  }
  // this lane's output column (p,q) -> voxel n
  int col = nt * 16 + l15;
  int p = col >> 3, q = col & 7;
  int wp = p % WP_, hp = (p / WP_) % HP_, sp = p / (WP_ * HP_);
  int dsq = q >> 2, dh = (q >> 1) & 1, dw = q & 1;
  int n = ((sp * 2 + dsq) * H_ + (hp * 2 + dh)) * W_ + (wp * 2 + dw);
  float* Yo = Yout + (size_t)b * C_ * N_;
#pragma unroll
  for (int rr = 0; rr < 8; rr++) {
    int m = mt * 16 + rr + half * 8;
    Yo[(size_t)m * N_ + n] = acc[rr] + bias[m];
  }
}

// ---------------------------------------------------------------------------
// per-row mean / rstd  (instance-norm statistics, biased variance)
// ---------------------------------------------------------------------------
__global__ __launch_bounds__(256)
void k_stats(const float* __restrict__ src, float* __restrict__ stats, int cols)
{
  int row = blockIdx.x;
  const float* pr = src + (size_t)row * cols;
  float s = 0.f, ss = 0.f;
  for (int i = threadIdx.x; i < cols; i += 256) { float v = pr[i]; s += v; ss += v * v; }
  __shared__ float sh0[256], sh1[256];
  int tid = threadIdx.x;
  sh0[tid] = s; sh1[tid] = ss; __syncthreads();
  for (int off = 128; off > 0; off >>= 1) {
    if (tid < off) { sh0[tid] += sh0[tid + off]; sh1[tid] += sh1[tid + off]; }
    __syncthreads();
  }
  if (tid == 0) {
    float mu = sh0[0] / cols;
    float var = sh1[0] / cols - mu * mu;
    stats[2 * row] = mu;
    stats[2 * row + 1] = rsqrtf(var + EPS_);
  }
}

// ---------------------------------------------------------------------------
// normalize (fc1 norm) + MaxPool3d(2) with within-block argmax
// ---------------------------------------------------------------------------
__global__ __launch_bounds__(256)
void k_pool(const float* __restrict__ y1, const float* __restrict__ stats,
            const float* __restrict__ gamma, const float* __restrict__ beta,
            float* __restrict__ xp, int* __restrict__ idxp)
{
  int gid = blockIdx.x * 256 + threadIdx.x;
  if (gid >= B_ * C_ * NP_) return;
  int p = gid % NP_;
  int bc = gid / NP_;
  int c = bc % C_;
  float mu = stats[2 * bc], rs = stats[2 * bc + 1];
  float g = gamma[c], be = beta[c];
  int wp = p % WP_, hp = (p / WP_) % HP_, sp = p / (WP_ * HP_);
  const float* src = y1 + (size_t)bc * N_;
  float best = -3.4e38f; int bq = 0;
#pragma unroll
  for (int q = 0; q < 8; q++) {            // argmax tie-break: first max
    int ds = q >> 2, dh = (q >> 1) & 1, dw = q & 1;
    int n = ((sp * 2 + ds) * H_ + (hp * 2 + dh)) * W_ + (wp * 2 + dw);
    float v = (src[n] - mu) * rs * g + be;
    if (v > best) { best = v; bq = q; }
  }
  xp[gid] = best; idxp[gid] = bq;
}

// ---------------------------------------------------------------------------
// channel l2-normalization of pooled node features; also emit squared norm
// ---------------------------------------------------------------------------
__global__ __launch_bounds__(256)
void k_l2norm(const float* __restrict__ xp, float* __restrict__ xn,
              float* __restrict__ sq)
{
  int gid = blockIdx.x * 256 + threadIdx.x;
  if (gid >= B_ * NP_) return;
  int b = gid / NP_, p = gid % NP_;
  const float* src = xp + (size_t)b * C_ * NP_ + p;
  float ss = 0.f;
  for (int c = 0; c < C_; c++) { float v = src[(size_t)c * NP_]; ss += v * v; }
  float r = rsqrtf(ss + 1e-12f);
  float* dst = xn + (size_t)b * C_ * NP_ + p;
  for (int c = 0; c < C_; c++) dst[(size_t)c * NP_] = src[(size_t)c * NP_] * r;
  sq[gid] = ss * r * r;
}

// ---------------------------------------------------------------------------
// KNN: one wave per 16-row tile.  A panel (128 x 16 rows) async-staged into
// LDS once; WMMA Gram 16x16 tiles -> LDS -> per-lane sorted top-9 insertion
// (ascending dist; ties keep lowest index like top_k).
// ---------------------------------------------------------------------------
__global__ __launch_bounds__(32)
void k_knn(const float* __restrict__ xn, const float* __restrict__ sq,
           int* __restrict__ nn)
{
  const int RT = NP_ / 16;                  // 288 row tiles
  int b = blockIdx.x / RT;
  int rt = blockIdx.x % RT;
  int n0 = rt * 16;
  const float* X = xn + (size_t)b * C_ * NP_;
  const float* Sq = sq + (size_t)b * NP_;
  int lane = threadIdx.x & 31;
  int half = lane >> 4, l15 = lane & 15;
  __shared__ __align__(16) float apanel[C_ * 16];   // [k][row]
  __shared__ float tile[16][17];

  // stage A panel: 128 rows of 16 floats = 512 b128 transfers
  for (int i = lane; i < C_ * 4; i += 32) {
    int row = i >> 2, cg = i & 3;
    g2l_async_b128(&apanel[row * 16 + cg * 4], X + (size_t)row * NP_ + n0 + cg * 4);
  }
  wait_async0();                            // single-wave block: no barrier needed

  float bd[KNN_]; int bi[KNN_];
#pragma unroll
  for (int t = 0; t < KNN_; t++) { bd[t] = 3.4e38f; bi[t] = 0; }
  float sn = (lane < 16) ? Sq[n0 + lane] : 0.f;

  for (int m0 = 0; m0 < NP_; m0 += 16) {
    v8f acc = {};
    for (int k = 0; k < C_; k += 4) {
      int kk = k + half * 2;
      v2f a, bf;
      a.x  = apanel[kk * 16 + l15];
      a.y  = apanel[(kk + 1) * 16 + l15];
      bf.x = X[(size_t)kk * NP_ + m0 + l15];
      bf.y = X[(size_t)(kk + 1) * NP_ + m0 + l15];
      acc = __builtin_amdgcn_wmma_f32_16x16x4_f32(false, a, false, bf,
                                                  (short)0, acc, false, false);
    }
#pragma unroll
    for (int rr = 0; rr < 8; rr++) tile[rr + half * 8][l15] = acc[rr];
    __syncthreads();
    if (lane < 16) {
#pragma unroll
      for (int j = 0; j < 16; j++) {
        int m = m0 + j;
        float d = sn + Sq[m] - 2.f * tile[lane][j];
        float pd = d; int pi = m;           // swap-cascade insertion
#pragma unroll
        for (int t = 0; t < KNN_; t++) {
          if (pd < bd[t]) {
            float td = bd[t]; int ti = bi[t];
            bd[t] = pd; bi[t] = pi; pd = td; pi = ti;
          }
        }
      }
    }
    __syncthreads();
  }
  if (lane < 16) {
    int* dst = nn + ((size_t)b * NP_ + n0 + lane) * KNN_;
#pragma unroll
    for (int t = 0; t < KNN_; t++) dst[t] = bi[t];
  }
}

// ---------------------------------------------------------------------------
// MRConv: gather K neighbors, max-relative; interleave [x, mr] channels
// ---------------------------------------------------------------------------
__global__ __launch_bounds__(256)
void k_mrconv(const float* __restrict__ xp, const int* __restrict__ nn,
              float* __restrict__ xcat)
{
  int gid = blockIdx.x * 256 + threadIdx.x;
  if (gid >= B_ * C_ * NP_) return;
  int p = gid % NP_;
  int bc = gid / NP_;
  int b = bc / C_, c = bc % C_;
  const float* xpb = xp + (size_t)b * C_ * NP_ + (size_t)c * NP_;
  const int* nnb = nn + ((size_t)b * NP_ + p) * KNN_;
  float x0 = xpb[p];
  float mr = -3.4e38f;
#pragma unroll
  for (int t = 0; t < KNN_; t++) {
    float dv = xpb[nnb[t]] - x0;
    mr = dv > mr ? dv : mr;
  }
  float* dst = xcat + (size_t)b * C2_ * NP_;
  dst[(size_t)(2 * c) * NP_ + p] = x0;
  dst[(size_t)(2 * c + 1) * NP_ + p] = mr;
}

// ---------------------------------------------------------------------------
// instance norm + leaky relu (in place)
// ---------------------------------------------------------------------------
__global__ __launch_bounds__(256)
void k_norm_lrelu(float* __restrict__ y, const float* __restrict__ stats,
                  const float* __restrict__ gamma, const float* __restrict__ beta)
{
  int gid = blockIdx.x * 256 + threadIdx.x;
  if (gid >= B_ * C2_ * NP_) return;
  int row = gid / NP_;
  int ch = row % C2_;
  float mu = stats[2 * row], rs = stats[2 * row + 1];
  float v = (y[gid] - mu) * rs * gamma[ch] + beta[ch];
  y[gid] = v >= 0.f ? v : SLOPE_ * v;
}

// ---------------------------------------------------------------------------
// final instance norm + residual (in place on d_out)
// ---------------------------------------------------------------------------
__global__ __launch_bounds__(256)
void k_final(float* __restrict__ out, const float* __restrict__ stats,
             const float* __restrict__ gamma, const float* __restrict__ beta,
             const float* __restrict__ xin)
{
  int gid = blockIdx.x * 256 + threadIdx.x;
  if (gid >= B_ * C_ * N_) return;
  int row = gid / N_;
  int c = row % C_;
  float mu = stats[2 * row], rs = stats[2 * row + 1];
  out[gid] = (out[gid] - mu) * rs * gamma[c] + beta[c] + xin[gid];
}

// ---------------------------------------------------------------------------
extern "C" void kernel_launch(void* const* d_in, const int* in_sizes, int n_in,
                              void* d_out, int out_size, void* d_ws, size_t ws_size,
                              hipStream_t stream)
{
  (void)in_sizes; (void)n_in; (void)out_size; (void)ws_size;
  const float* x      = (const float*)d_in[0];
  const float* fc1_w  = (const float*)d_in[1];
  const float* fc1_b  = (const float*)d_in[2];
  const float* fc1_g  = (const float*)d_in[3];
  const float* fc1_be = (const float*)d_in[4];
  const float* g_w    = (const float*)d_in[5];
  const float* g_b    = (const float*)d_in[6];
  const float* g_g    = (const float*)d_in[7];
  const float* g_be   = (const float*)d_in[8];
  const float* fc2_w  = (const float*)d_in[9];
  const float* fc2_b  = (const float*)d_in[10];
  const float* fc2_g  = (const float*)d_in[11];
  const float* fc2_be = (const float*)d_in[12];
  float* out = (float*)d_out;

  char* base = (char*)d_ws;
  size_t off = 0;
  auto alloc = [&](size_t bytes) -> char* {
    char* p = base + off; off += (bytes + 255) & ~(size_t)255; return p;
  };
  float* y1   = (float*)alloc((size_t)B_ * C_ * N_ * 4);    // fc1 raw; later aliased
  float* xp   = (float*)alloc((size_t)B_ * C_ * NP_ * 4);
  int*   idxp = (int*)  alloc((size_t)B_ * C_ * NP_ * 4);
  float* xn   = (float*)alloc((size_t)B_ * C_ * NP_ * 4);
  float* sqv  = (float*)alloc((size_t)B_ * NP_ * 4);
  int*   nn   = (int*)  alloc((size_t)B_ * NP_ * KNN_ * 4);
  float* st1  = (float*)alloc((size_t)B_ * C_ * 2 * 4);
  float* st2  = (float*)alloc((size_t)B_ * C2_ * 2 * 4);
  float* st3  = (float*)alloc((size_t)B_ * C_ * 2 * 4);
  float* xcat = y1;                                // y1 dead after k_pool
  float* y2   = y1 + (size_t)B_ * C2_ * NP_;

  // 1. fc1 1x1-conv GEMM (raw), LDS-staged panels
  k_gemm_panel<C_, C_><<<B_ * (N_ / 16), 256, 0, stream>>>(fc1_w, fc1_b, x, y1, N_);
  // 2. fc1 instance-norm stats
  k_stats<<<B_ * C_, 256, 0, stream>>>(y1, st1, N_);
  // 3. normalize + maxpool with argmax
  k_pool<<<(B_ * C_ * NP_ + 255) / 256, 256, 0, stream>>>(
      y1, st1, fc1_g, fc1_be, xp, idxp);
  // 4. channel l2 norm
  k_l2norm<<<(B_ * NP_ + 255) / 256, 256, 0, stream>>>(xp, xn, sqv);
  // 5. dense KNN via WMMA Gram tiles + fused top-9
  k_knn<<<B_ * (NP_ / 16), 32, 0, stream>>>(xn, sqv, nn);
  // 6. MRConv gather / max-relative / interleave
  k_mrconv<<<(B_ * C_ * NP_ + 255) / 256, 256, 0, stream>>>(xp, nn, xcat);
  // 7. graph 1x1-conv GEMM (2C->2C, raw), LDS-staged panels
  k_gemm_panel<C2_, C2_><<<B_ * (NP_ / 16), 256, 0, stream>>>(g_w, g_b, xcat, y2, NP_);
  // 8. stats, 9. norm + leaky relu
  k_stats<<<B_ * C2_, 256, 0, stream>>>(y2, st2, NP_);
  k_norm_lrelu<<<(B_ * C2_ * NP_ + 255) / 256, 256, 0, stream>>>(
      y2, st2, g_g, g_be);
  // 10. fc2 GEMM with fused unpool (panel style) -> raw output in d_out
  k_gemm_fc2<<<B_ * (N_ / 16), 256, 0, stream>>>(fc2_w, fc2_b, y2, idxp, out);
  // 11. stats, 12. normalize + residual in place
  k_stats<<<B_ * C_, 256, 0, stream>>>(out, st3, N_);
  k_final<<<(B_ * C_ * N_ + 255) / 256, 256, 0, stream>>>(
      out, st3, fc2_g, fc2_be, x);
}